// OuterAttention_63848983822768
// MI455X (gfx1250) — compile-verified
//
#include <hip/hip_runtime.h>
#include <hip/hip_bf16.h>
#include <math.h>

// Problem dims (fixed by the reference)
#define NA   8192   // rows of seq1 (a)
#define NB   8192   // rows of seq2 (b)
#define NE   1024   // e1 == e2 == c == m == 1024
#define NM   1024

typedef __attribute__((ext_vector_type(2))) float v2f;
typedef __attribute__((ext_vector_type(8))) float v8f;

// ---------------- workspace layout (floats) ----------------
#define OFF_P1   ((size_t)0)
#define OFF_P2   ((size_t)NA * NM)                 // 8388608
#define OFF_CTX1 (OFF_P2 + (size_t)NB * NM)        // 16777216
#define OFF_CTX2 (OFF_CTX1 + NM)
#define OFF_S1   (OFF_CTX2 + NM)                   // s1,s2 contiguous (2048)
#define OFF_S2   (OFF_S1 + NM)
#define OFF_LA   (OFF_S2 + NM)
#define OFF_LB   (OFF_LA + NA)
#define OFF_WA   (OFF_LB + NB)
#define OFF_WB   (OFF_WA + NA)

__device__ __forceinline__ float waveReduceSum(float v) {
    #pragma unroll
    for (int m = 16; m > 0; m >>= 1) v += __shfl_xor(v, m, 32);
    return v;
}

// ---------------- 0: zero init (d_out + column-sum accumulators) -------------
__global__ __launch_bounds__(256) void zero_kernel(float* out2048, float* s2048) {
    int i = blockIdx.x * 256 + threadIdx.x;   // grid: 8 blocks -> 2048 threads
    if (i < 2 * NE) { out2048[i] = 0.0f; s2048[i] = 0.0f; }
}

// ---------------- 1: ctx projections: ctx = Wc @ c + bias -------------------
// 2048 rows total (1024 per side); one wave per row.
__global__ __launch_bounds__(256) void ctx_kernel(const float* __restrict__ Wc1,
                                                  const float* __restrict__ b1,
                                                  const float* __restrict__ Wc2,
                                                  const float* __restrict__ b2,
                                                  const float* __restrict__ cvec,
                                                  float* __restrict__ ctx1,
                                                  float* __restrict__ ctx2) {
    int g    = blockIdx.x * 8 + (threadIdx.x >> 5);  // global row id 0..2047
    int lane = threadIdx.x & 31;
    int side = g >> 10;
    int row  = g & (NM - 1);
    const float* W    = side ? Wc2 : Wc1;
    const float* bias = side ? b2  : b1;
    float*       o    = side ? ctx2 : ctx1;
    float acc = 0.0f;
    for (int k = lane; k < NE; k += 32)
        acc += W[(size_t)row * NE + k] * cvec[k];
    acc = waveReduceSum(acc);
    if (lane == 0) o[row] = acc + bias[row];
}

// ---------------- 2: fused GEMM + ctx + ELU via V_WMMA_F32_16X16X4_F32 ------
// P[r, n] = elu( sum_k X[r,k] * W[n,k] + ctx[n] ),  X: [NA x NE], W: [NM x NE]
// One wave computes a 16(M) x 64(N) tile: one A fragment is reused against four
// B fragments (4 accumulators) -> 5 loads per 4 WMMAs instead of 8.
// A (16x4, MxK): lanes 0-15 hold M=lane, K={k0,k0+1}; lanes 16-31 K={k0+2,k0+3}
// B (4x16, KxN): lanes 0-15 hold N=lane, K={k0,k0+1}; lanes 16-31 K={k0+2,k0+3}
//   and B[k][n] = W[n][k]  -> identical addressing pattern as A over W's row n.
// C/D (16x16): vgpr j, lanes 0-15 -> (M=j, N=lane); lanes 16-31 -> (M=j+8, N=lane-16)
__global__ __launch_bounds__(256) void gemm_elu_kernel(const float* __restrict__ X,
                                                       const float* __restrict__ W,
                                                       const float* __restrict__ ctx,
                                                       float* __restrict__ P) {
    const int wave = threadIdx.x >> 5;
    const int lane = threadIdx.x & 31;
    const int tile  = blockIdx.x * 8 + wave;         // 8192 tiles of 16x64
    const int tileM = tile >> 4;                     // 512 M-tiles
    const int tileN = tile & 15;                     // 16 N-supertiles (64 wide)
    const int row0 = tileM << 4;
    const int col0 = tileN << 6;

    const int half  = lane & 15;                     // A: M index, B: N index
    const int khalf = (lane >> 4) << 1;              // 0 or 2

    const float* aptr = X + (size_t)(row0 + half) * NE + khalf;
    const float* bptr = W + (size_t)(col0 + half) * NE + khalf;

    v8f acc0 = {0.f,0.f,0.f,0.f,0.f,0.f,0.f,0.f};
    v8f acc1 = acc0, acc2 = acc0, acc3 = acc0;

    #pragma unroll 4
    for (int k = 0; k < NE; k += 4) {
        v2f av  = *(const v2f*)(aptr + k);
        v2f bv0 = *(const v2f*)(bptr + k);
        v2f bv1 = *(const v2f*)(bptr + 16 * NE + k);
        v2f bv2 = *(const v2f*)(bptr + 32 * NE + k);
        v2f bv3 = *(const v2f*)(bptr + 48 * NE + k);
        acc0 = __builtin_amdgcn_wmma_f32_16x16x4_f32(false, av, false, bv0, (short)0, acc0, false, false);
        acc1 = __builtin_amdgcn_wmma_f32_16x16x4_f32(false, av, false, bv1, (short)0, acc1, false, false);
        acc2 = __builtin_amdgcn_wmma_f32_16x16x4_f32(false, av, false, bv2, (short)0, acc2, false, false);
        acc3 = __builtin_amdgcn_wmma_f32_16x16x4_f32(false, av, false, bv3, (short)0, acc3, false, false);
    }

    const int rbase = row0 + ((lane >> 4) << 3);
    v8f accs[4] = {acc0, acc1, acc2, acc3};
    #pragma unroll
    for (int g = 0; g < 4; ++g) {
        const int colg = col0 + (g << 4) + half;
        const float cv = ctx[colg];
        #pragma unroll
        for (int j = 0; j < 8; ++j) {
            float x = accs[g][j] + cv;
            // branchless ELU (alpha = 1): fast exp only affects the negative tail
            x = (x > 0.0f) ? x : (__expf(x) - 1.0f);
            P[(size_t)(rbase + j) * NM + colg] = x;
        }
    }
}

// ---------------- 3: column sums s[n] = sum_r P[r,n] ------------------------
__global__ __launch_bounds__(256) void colsum_kernel(const float* __restrict__ P,
                                                     float* __restrict__ s) {
    int col = blockIdx.x * 256 + threadIdx.x;        // gridDim.x = 4
    int r0  = blockIdx.y * 256;                      // gridDim.y = 32
    float acc = 0.0f;
    for (int r = r0; r < r0 + 256; ++r)
        acc += P[(size_t)r * NM + col];
    unsafeAtomicAdd(&s[col], acc);
}

// ---------------- 4: logits[r] = dot(P[r,:], s) (wave per row) --------------
__global__ __launch_bounds__(256) void logits_kernel(const float* __restrict__ P,
                                                     const float* __restrict__ s,
                                                     float* __restrict__ logits) {
    int row  = blockIdx.x * 8 + (threadIdx.x >> 5);  // grid 1024 blocks -> 8192 rows
    int lane = threadIdx.x & 31;
    float acc = 0.0f;
    for (int k = lane; k < NM; k += 32)
        acc += P[(size_t)row * NM + k] * s[k];
    acc = waveReduceSum(acc);
    if (lane == 0) logits[row] = acc;
}

// ---------------- 5: softmax over 8192 logits (single block) ----------------
__global__ __launch_bounds__(1024) void softmax_kernel(const float* __restrict__ logits,
                                                       float* __restrict__ w, int n) {
    __shared__ float red[1024];
    int t = threadIdx.x;
    float mx = -INFINITY;
    for (int i = t; i < n; i += 1024) mx = fmaxf(mx, logits[i]);
    red[t] = mx; __syncthreads();
    for (int sft = 512; sft > 0; sft >>= 1) {
        if (t < sft) red[t] = fmaxf(red[t], red[t + sft]);
        __syncthreads();
    }
    mx = red[0]; __syncthreads();
    float sum = 0.0f;
    for (int i = t; i < n; i += 1024) sum += expf(logits[i] - mx);
    red[t] = sum; __syncthreads();
    for (int sft = 512; sft > 0; sft >>= 1) {
        if (t < sft) red[t] += red[t + sft];
        __syncthreads();
    }
    float inv = 1.0f / red[0];
    for (int i = t; i < n; i += 1024) w[i] = expf(logits[i] - mx) * inv;
}

// ---------------- 6: pooled output att[n] = sum_r w[r] * X[r,n] -------------
__global__ __launch_bounds__(256) void pool_kernel(const float* __restrict__ w,
                                                   const float* __restrict__ X,
                                                   float* __restrict__ out) {
    int col = blockIdx.x * 256 + threadIdx.x;        // gridDim.x = 4
    int r0  = blockIdx.y * 512;                      // gridDim.y = 16
    float acc = 0.0f;
    for (int r = r0; r < r0 + 512; ++r)
        acc += w[r] * X[(size_t)r * NE + col];
    unsafeAtomicAdd(&out[col], acc);
}

extern "C" void kernel_launch(void* const* d_in, const int* in_sizes, int n_in,
                              void* d_out, int out_size, void* d_ws, size_t ws_size,
                              hipStream_t stream) {
    const float* seq1 = (const float*)d_in[0];   // [8192, 1024]
    const float* seq2 = (const float*)d_in[1];   // [8192, 1024]
    const float* cvec = (const float*)d_in[2];   // [1024]
    const float* Wc1  = (const float*)d_in[3];   // [1024, 1024]
    const float* W1   = (const float*)d_in[4];   // [1024, 1024]
    const float* b1   = (const float*)d_in[5];   // [1024]
    const float* Wc2  = (const float*)d_in[6];   // [1024, 1024]
    const float* W2   = (const float*)d_in[7];   // [1024, 1024]
    const float* b2   = (const float*)d_in[8];   // [1024]
    float* out = (float*)d_out;                  // att1[1024] ++ att2[1024]
    float* ws  = (float*)d_ws;

    float* P1   = ws + OFF_P1;
    float* P2   = ws + OFF_P2;
    float* ctx1 = ws + OFF_CTX1;
    float* ctx2 = ws + OFF_CTX2;
    float* s1   = ws + OFF_S1;
    float* s2   = ws + OFF_S2;
    float* la   = ws + OFF_LA;
    float* lb   = ws + OFF_LB;
    float* wa   = ws + OFF_WA;
    float* wb   = ws + OFF_WB;

    // 0: zero d_out and s1/s2 (contiguous 2048 floats at s1)
    zero_kernel<<<8, 256, 0, stream>>>(out, s1);

    // 1: context projections (both sides in one launch)
    ctx_kernel<<<256, 256, 0, stream>>>(Wc1, b1, Wc2, b2, cvec, ctx1, ctx2);

    // 2: projection GEMMs + ELU (fp32 WMMA), 8192 wave-tiles of 16x64
    gemm_elu_kernel<<<1024, 256, 0, stream>>>(seq1, W1, ctx1, P1);
    gemm_elu_kernel<<<1024, 256, 0, stream>>>(seq2, W2, ctx2, P2);

    // 3: column sums of P1 / P2
    colsum_kernel<<<dim3(4, 32), 256, 0, stream>>>(P1, s1);
    colsum_kernel<<<dim3(4, 32), 256, 0, stream>>>(P2, s2);

    // 4: logits via the outer-product-sum identity
    logits_kernel<<<1024, 256, 0, stream>>>(P1, s2, la);
    logits_kernel<<<1024, 256, 0, stream>>>(P2, s1, lb);

    // 5: softmax weights
    softmax_kernel<<<1, 1024, 0, stream>>>(la, wa, NA);
    softmax_kernel<<<1, 1024, 0, stream>>>(lb, wb, NB);

    // 6: weighted pooling into d_out
    pool_kernel<<<dim3(4, 16), 256, 0, stream>>>(wa, seq1, out);
    pool_kernel<<<dim3(4, 16), 256, 0, stream>>>(wb, seq2, out + NE);
}